// ECELoss_87119116632190
// MI455X (gfx1250) — compile-verified
//
#include <hip/hip_runtime.h>
#include <hip/hip_bf16.h>
#include <stdint.h>

#define NBINS 15

typedef float v4f __attribute__((ext_vector_type(4)));
typedef int   v4i __attribute__((vector_size(16)));          // matches builtin param pointee
typedef __attribute__((address_space(1))) v4i g_v4i;          // global (AS1)
typedef __attribute__((address_space(3))) v4i l_v4i;          // LDS (AS3)

// CDNA5 async global->LDS path (gfx1250): GLOBAL_LOAD_ASYNC_TO_LDS_B128 + S_WAIT_ASYNCCNT.
#if defined(__gfx1250__) && __has_builtin(__builtin_amdgcn_global_load_async_to_lds_b128) && \
    __has_builtin(__builtin_amdgcn_s_wait_asynccnt)
#define USE_ASYNC_LDS 1
#else
#define USE_ASYNC_LDS 0
#endif

__device__ __forceinline__ void online_update(float x, int idx, float& m, float& s, int& am) {
    // Online softmax-max: keep running max m, running sum of exp(x - m), first argmax.
    if (x > m) {
        s = s * __expf(m - x) + 1.0f;  // rescale old sum, add exp(x - x) = 1
        m = x;
        am = idx;
    } else {
        s += __expf(x - m);
    }
}

__global__ void __launch_bounds__(256) ece_init(float* gbins) {
    int t = threadIdx.x;
    if (t < 3 * NBINS) gbins[t] = 0.0f;
}

__global__ void __launch_bounds__(256) ece_partial(const float* __restrict__ logits,
                                                   const int* __restrict__ labels,
                                                   float* __restrict__ gbins,
                                                   int N, int C4) {
    __shared__ float sbins[3 * NBINS];
#if USE_ASYNC_LDS
    __shared__ v4f stage[8][4][32];  // [wave][buf][lane] : 16 KB, 4-deep async ring
#endif
    const int tid  = threadIdx.x;
    const int lane = tid & 31;  // wave32
    const int wave = tid >> 5;

    if (tid < 3 * NBINS) sbins[tid] = 0.0f;
    __syncthreads();

    const int wavesPerBlock = blockDim.x >> 5;
    const int nwaves        = gridDim.x * wavesPerBlock;

    for (int row = blockIdx.x * wavesPerBlock + wave; row < N; row += nwaves) {
        const v4f* __restrict__ rowp = (const v4f*)logits + (size_t)row * (size_t)C4;

        float m  = -3.0e38f;
        float s  = 0.0f;
        int   am = 0;

        const int nChunks = (C4 + 31) >> 5;

#if USE_ASYNC_LDS
        // Prologue: stage chunks 0..2 into the ring (one b128 per lane, wave-private).
        for (int k = 0; k < 3 && k < nChunks; ++k) {
            const int i = (k << 5) + lane;
            if (i < C4) {
                __builtin_amdgcn_global_load_async_to_lds_b128(
                    (g_v4i*)(rowp + i),
                    (l_v4i*)&stage[wave][k & 3][lane],
                    0, 0);
            }
        }
        for (int c = 0; c < nChunks; ++c) {
            const int buf = c & 3;
            const int cnext = c + 3;
            if (cnext < nChunks) {
                const int i = (cnext << 5) + lane;
                if (i < C4) {
                    __builtin_amdgcn_global_load_async_to_lds_b128(
                        (g_v4i*)(rowp + i),
                        (l_v4i*)&stage[wave][cnext & 3][lane],
                        0, 0);
                }
            }
            // Wait until chunk c has landed: allow min(3, remaining) still in flight.
            const int rem = nChunks - 1 - c;
            if (rem >= 3)      __builtin_amdgcn_s_wait_asynccnt(3);
            else if (rem == 2) __builtin_amdgcn_s_wait_asynccnt(2);
            else if (rem == 1) __builtin_amdgcn_s_wait_asynccnt(1);
            else               __builtin_amdgcn_s_wait_asynccnt(0);

            const int i = (c << 5) + lane;
            if (i < C4) {
                v4f v = stage[wave][buf][lane];
                const int base = i << 2;
                online_update(v.x, base + 0, m, s, am);
                online_update(v.y, base + 1, m, s, am);
                online_update(v.z, base + 2, m, s, am);
                online_update(v.w, base + 3, m, s, am);
            }
        }
#else
        for (int c = 0; c < nChunks; ++c) {
            const int i = (c << 5) + lane;
            if (i + 32 < C4) __builtin_prefetch(&rowp[i + 32], 0, 0);  // global_prefetch_b8
            if (i < C4) {
                v4f v = __builtin_nontemporal_load(rowp + i);  // stream-once data
                const int base = i << 2;
                online_update(v.x, base + 0, m, s, am);
                online_update(v.y, base + 1, m, s, am);
                online_update(v.z, base + 2, m, s, am);
                online_update(v.w, base + 3, m, s, am);
            }
        }
#endif

        // Wave32 tree reduction of (max, rescaled sum, first-argmax).
        for (int off = 16; off > 0; off >>= 1) {
            float m2 = __shfl_xor(m, off, 32);
            float s2 = __shfl_xor(s, off, 32);
            int   a2 = __shfl_xor(am, off, 32);
            float mn = fmaxf(m, m2);
            float sn = s * __expf(m - mn) + s2 * __expf(m2 - mn);
            if (m2 > m || (m2 == m && a2 < am)) am = a2;
            m = mn;
            s = sn;
        }

        if (lane == 0) {
            float conf = 1.0f / s;  // exp(m - m) / sum exp(l - m)
            int bin = (int)ceilf(conf * (float)NBINS) - 1;
            bin = bin < 0 ? 0 : (bin > NBINS - 1 ? NBINS - 1 : bin);
            float acc = (am == labels[row]) ? 1.0f : 0.0f;
            atomicAdd(&sbins[bin], 1.0f);              // ds_add_f32
            atomicAdd(&sbins[NBINS + bin], conf);
            atomicAdd(&sbins[2 * NBINS + bin], acc);
        }
    }

    __syncthreads();
    if (tid < 3 * NBINS) atomicAdd(&gbins[tid], sbins[tid]);  // one flush per block
}

__global__ void __launch_bounds__(32) ece_final(const float* __restrict__ gbins,
                                                float* __restrict__ out, float invN) {
    int b = threadIdx.x;
    float term = 0.0f;
    if (b < NBINS) {
        float cnt = gbins[b];
        float cs  = gbins[NBINS + b];
        float as  = gbins[2 * NBINS + b];
        if (cnt > 0.0f) {
            term = fabsf(cs / cnt - as / cnt) * (cnt * invN);
        }
    }
    for (int off = 16; off > 0; off >>= 1) term += __shfl_xor(term, off, 32);
    if (b == 0) out[0] = term;
}

extern "C" void kernel_launch(void* const* d_in, const int* in_sizes, int n_in,
                              void* d_out, int out_size, void* d_ws, size_t ws_size,
                              hipStream_t stream) {
    const float* logits = (const float*)d_in[0];
    const int*   labels = (const int*)d_in[1];

    const int N  = in_sizes[1];               // 262144 samples
    const int C  = in_sizes[0] / in_sizes[1]; // 1000 classes (divisible by 4)
    const int C4 = C >> 2;                    // 250 float4 per row

    float* gbins = (float*)d_ws;              // 45 floats: counts | conf_sum | acc_sum

    ece_init<<<1, 64, 0, stream>>>(gbins);

    const int blocks = 4096;                  // 32768 waves -> 8 rows per wave
    ece_partial<<<blocks, 256, 0, stream>>>(logits, labels, gbins, N, C4);

    ece_final<<<1, 32, 0, stream>>>(gbins, (float*)d_out, 1.0f / (float)N);
}